// DDP_input_14637248545061
// MI455X (gfx1250) — compile-verified
//
#include <hip/hip_runtime.h>
#include <cstdint>

#ifndef __has_builtin
#define __has_builtin(x) 0
#endif

#if __has_builtin(__builtin_amdgcn_global_load_async_to_lds_b128)
#define HAVE_ASYNC_B128 1
#else
#define HAVE_ASYNC_B128 0
#endif

#if defined(__AMDGCN__) && !HAVE_ASYNC_B128
#warning "gfx1250 device pass: __builtin_amdgcn_global_load_async_to_lds_b128 NOT found; using VGPR staging fallback"
#endif

namespace {

constexpr int B = 32;
constexpr int C = 3;
constexpr int H = 512;
constexpr int W = 512;
constexpr int K = 15;     // window
constexpr int R = K / 2;  // 7

typedef int v4i __attribute__((vector_size(16)));
typedef __attribute__((address_space(1))) v4i* g_v4i_ptr;  // global (AS1)
typedef __attribute__((address_space(3))) v4i* l_v4i_ptr;  // LDS (AS3)

__device__ __forceinline__ float fmin3(float a, float b, float c) {
  return fminf(a, fminf(b, c));
}

__device__ __forceinline__ void async_copy_b128(const void* g, void* l) {
#if HAVE_ASYNC_B128
  __builtin_amdgcn_global_load_async_to_lds_b128(
      (g_v4i_ptr)(uintptr_t)g,
      (l_v4i_ptr)(uintptr_t)l,
      /*imm offset=*/0, /*cpol=*/0);
#else
  (void)g; (void)l;
#endif
}

__device__ __forceinline__ void wait_async_zero() {
#if __has_builtin(__builtin_amdgcn_s_wait_asynccnt)
  __builtin_amdgcn_s_wait_asynccnt(0);
#elif defined(__AMDGCN__)
  asm volatile("s_wait_asynccnt 0" ::: "memory");
#endif
}

// -------- Pass 1: channel-min (dark channel) + horizontal 15-tap min --------
// One block per (b, y) row. 128 threads, 4 pixels/thread.
__global__ __launch_bounds__(128) void dark_hmin_kernel(
    const float* __restrict__ I, float* __restrict__ hmin) {
  const int tid = threadIdx.x;          // 0..127
  const int row = blockIdx.x;           // b*H + y
  const int b = row >> 9;               // /H
  const int y = row & (H - 1);

  __shared__ __align__(16) float sc[C][W];        // staged channel rows (6 KB)
  __shared__ __align__(16) float sd[8 + W + 8];   // dark row + halo, dark at +8

  const float PINF = __int_as_float(0x7f800000);
  const float* rbase = I + (((size_t)b * C) * H + (size_t)y) * W;  // channel 0 row

#if HAVE_ASYNC_B128
  {
    const int byteOff = tid * 16;  // 128 lanes * 16 B = 2048 B = one 512-float row
#pragma unroll
    for (int c = 0; c < C; ++c) {
      async_copy_b128((const char*)(rbase + (size_t)c * H * W) + byteOff,
                      (char*)(&sc[c][0]) + byteOff);
    }
    wait_async_zero();
  }
#else
#pragma unroll
  for (int c = 0; c < C; ++c) {
    reinterpret_cast<float4*>(&sc[c][0])[tid] =
        reinterpret_cast<const float4*>(rbase + (size_t)c * H * W)[tid];
  }
#endif

  if (tid < 8) {  // +inf halo (torch min-pool pads with +inf)
    sd[tid] = PINF;
    sd[8 + W + tid] = PINF;
  }
  __syncthreads();

  {  // dark channel, 4 px/thread via float4 LDS traffic
    const float4 r = reinterpret_cast<const float4*>(&sc[0][0])[tid];
    const float4 g = reinterpret_cast<const float4*>(&sc[1][0])[tid];
    const float4 u = reinterpret_cast<const float4*>(&sc[2][0])[tid];
    float4 d;
    d.x = fmin3(r.x, g.x, u.x);
    d.y = fmin3(r.y, g.y, u.y);
    d.z = fmin3(r.z, g.z, u.z);
    d.w = fmin3(r.w, g.w, u.w);
    reinterpret_cast<float4*>(&sd[8])[tid] = d;
  }
  __syncthreads();

  // Horizontal 15-tap min for 4 consecutive outputs.
  // Output x0+i uses dark[x0+i-7 .. x0+i+7] = sd[x0+1+i .. x0+15+i] = v[i..i+14].
  const int x0 = tid * 4;
  float v[K + 3];
#pragma unroll
  for (int j = 0; j < K + 3; ++j) v[j] = sd[x0 + 1 + j];

  float suf[K];  // suf[i] = min(v[i..14])
  suf[K - 1] = v[K - 1];
#pragma unroll
  for (int j = K - 2; j >= 0; --j) suf[j] = fminf(v[j], suf[j + 1]);

  float4 o;
  o.x = suf[0];
  float q = v[K];             // v[15]
  o.y = fminf(suf[1], q);
  q = fminf(q, v[K + 1]);     // min(v[15..16])
  o.z = fminf(suf[2], q);
  q = fminf(q, v[K + 2]);     // min(v[15..17])
  o.w = fminf(suf[3], q);

  reinterpret_cast<float4*>(hmin + ((size_t)b * H + (size_t)y) * W)[tid] = o;
}

// -------- Pass 2: vertical 15-tap min, 8 output rows per thread --------
__global__ __launch_bounds__(256) void vmin_kernel(
    const float* __restrict__ hmin, float* __restrict__ out) {
  constexpr int TY = 8;
  const int x = blockIdx.x * blockDim.x + threadIdx.x;  // 0..511, coalesced
  const int y0 = blockIdx.y * TY;
  const int b = blockIdx.z;

  const float PINF = __int_as_float(0x7f800000);
  const float* col = hmin + ((size_t)b * H) * W + x;

  float v[TY + K - 1];  // rows y0-7 .. y0+14 (22 values)
#pragma unroll
  for (int j = 0; j < TY + K - 1; ++j) {
    const int yy = y0 - R + j;
    v[j] = (yy >= 0 && yy < H) ? col[(size_t)yy * W] : PINF;
  }

  float suf[K];  // suf[i] = min(v[i..14]); output i = min(v[i..i+14])
  suf[K - 1] = v[K - 1];
#pragma unroll
  for (int j = K - 2; j >= 0; --j) suf[j] = fminf(v[j], suf[j + 1]);

  float* obase = out + ((size_t)b * H + (size_t)y0) * W + x;
  float q = PINF;
#pragma unroll
  for (int i = 0; i < TY; ++i) {
    float m;
    if (i == 0) {
      m = suf[0];
    } else {
      q = fminf(q, v[K - 1 + i]);  // running min of v[15 .. 14+i]
      m = fminf(suf[i], q);
    }
    obase[(size_t)i * W] = m;
  }
}

}  // namespace

extern "C" void kernel_launch(void* const* d_in, const int* in_sizes, int n_in,
                              void* d_out, int out_size, void* d_ws, size_t ws_size,
                              hipStream_t stream) {
  (void)in_sizes; (void)n_in; (void)out_size; (void)ws_size;
  const float* I = (const float*)d_in[0];   // [32,3,512,512] fp32; k==15 fixed
  float* out = (float*)d_out;               // [32,1,512,512] fp32
  float* ws = (float*)d_ws;                 // 32*512*512 floats = 32 MiB scratch

  dim3 grid1(B * H);                        // 16384 row blocks
  dark_hmin_kernel<<<grid1, 128, 0, stream>>>(I, ws);

  dim3 grid2(W / 256, H / 8, B);            // 2 x 64 x 32 blocks
  vmin_kernel<<<grid2, 256, 0, stream>>>(ws, out);
}